// HypergraphLayer_3650722201951
// MI455X (gfx1250) — compile-verified
//
#include <hip/hip_runtime.h>
#include <hip/hip_bf16.h>

typedef __attribute__((ext_vector_type(2))) float v2f;
typedef __attribute__((ext_vector_type(8))) float v8f;

#define NEG_SLOPE 0.01f
#define DFEAT 64
#define GMAX  64

// ---------------------------------------------------------------------------
// Kernel 1: zero the workspace accumulators (g_sum[G*64] + counts[G]).
// ---------------------------------------------------------------------------
__global__ void hg_zero_kernel(float* __restrict__ ws, int n) {
    int i = blockIdx.x * blockDim.x + threadIdx.x;
    if (i < n) ws[i] = 0.0f;
}

// ---------------------------------------------------------------------------
// Kernel 2: counts[g] = # nodes with graph_ids == g.  LDS histogram per block,
// one global atomic per bin per block.
// ---------------------------------------------------------------------------
__global__ void hg_count_kernel(const int* __restrict__ graph_ids,
                                float* __restrict__ counts, int N, int G) {
    __shared__ float bins[GMAX];
    if (threadIdx.x < GMAX) bins[threadIdx.x] = 0.0f;
    __syncthreads();
    for (int i = blockIdx.x * blockDim.x + threadIdx.x; i < N;
         i += gridDim.x * blockDim.x) {
        atomicAdd(&bins[graph_ids[i]], 1.0f);   // ds_add_f32
    }
    __syncthreads();
    if (threadIdx.x < G) atomicAdd(&counts[threadIdx.x], bins[threadIdx.x]);
}

// ---------------------------------------------------------------------------
// Kernel 3: fused edge gather + scale + per-graph reduction.
//   g_sum[gid(dst[e])][:] += x[src[e]][:] * node_norm[src]*node_norm[dst]*edge_norm[e]
// Each wave32 processes 2 edges per iteration: 16 lanes per edge, one float4
// (16B) of the 256B feature row per lane.  Accumulate into a 64x64 f32 LDS
// tile with ds_add_f32, flush once per block with global_atomic_add_f32.
// ---------------------------------------------------------------------------
__global__ void hg_edge_kernel(const float* __restrict__ x,
                               const float* __restrict__ node_norm,
                               const float* __restrict__ edge_norm,
                               const int*   __restrict__ src,
                               const int*   __restrict__ dst,
                               const int*   __restrict__ graph_ids,
                               float* __restrict__ g_sum,
                               int E, int G) {
    __shared__ float acc[GMAX * DFEAT];        // 16 KB of the 320 KB/WGP LDS
    for (int i = threadIdx.x; i < GMAX * DFEAT; i += blockDim.x) acc[i] = 0.0f;
    __syncthreads();

    const int lane  = threadIdx.x & 31;
    const int chunk = lane & 15;               // which float4 of the row
    const int sub   = lane >> 4;               // edge 0/1 within the wave
    const int waveId = (blockIdx.x * blockDim.x + threadIdx.x) >> 5;
    const int nWaves = (gridDim.x * blockDim.x) >> 5;

    for (int e = waveId * 2 + sub; e < E; e += nWaves * 2) {
        const int s = src[e];
        const int d = dst[e];
        const float coef = node_norm[s] * node_norm[d] * edge_norm[e];
        const int g = graph_ids[d];
        const float4 v = ((const float4*)(x + (size_t)s * DFEAT))[chunk];
        float* a = &acc[g * DFEAT + chunk * 4];
        atomicAdd(a + 0, v.x * coef);          // ds_add_f32 x4
        atomicAdd(a + 1, v.y * coef);
        atomicAdd(a + 2, v.z * coef);
        atomicAdd(a + 3, v.w * coef);
    }
    __syncthreads();
    for (int i = threadIdx.x; i < G * DFEAT; i += blockDim.x)
        atomicAdd(&g_sum[i], acc[i]);          // global_atomic_add_f32
}

// ---------------------------------------------------------------------------
// Kernel 4: g_mean = g_sum / max(counts,1);  y = LeakyReLU(g_mean @ W^T + b)
// One 512-thread block = 16 waves; wave w owns output tile (w/4, w%4).
// 16 x V_WMMA_F32_16X16X4_F32 per wave over the K=64 dimension (fp32 WMMA,
// full reference precision).
//
// ISA f32 fragment layouts (CDNA5 ISA):
//   A 16x4 : lanes 0-15  -> M=lane,    v[0]=K0, v[1]=K1
//            lanes 16-31 -> M=lane-16, v[0]=K2, v[1]=K3
//   B 4x16 : same striping with N in place of M.
//   C/D    : lanes 0-15 N=lane (M=r in v[r]); lanes 16-31 N=lane-16 (M=r+8).
// ---------------------------------------------------------------------------
__global__ void hg_gemm_kernel(const float* __restrict__ g_sum,
                               const float* __restrict__ counts,
                               const float* __restrict__ W,
                               const float* __restrict__ bias,
                               float* __restrict__ y, int G) {
    __shared__ float gm[GMAX * DFEAT];
    for (int i = threadIdx.x; i < GMAX * DFEAT; i += blockDim.x) {
        int g = i >> 6;
        float c = (g < G) ? counts[g] : 1.0f;
        float s = (g < G) ? g_sum[i] : 0.0f;
        gm[i] = s / fmaxf(c, 1.0f);
    }
    __syncthreads();

    const int wave = threadIdx.x >> 5;         // 0..15
    const int lane = threadIdx.x & 31;
    const int tm = wave >> 2;                  // output row tile (graphs)
    const int tn = wave & 3;                   // output col tile (d_out)
    const int half = lane >> 4;
    const int idx  = lane & 15;

    v8f cacc = {};
    #pragma unroll
    for (int k0 = 0; k0 < DFEAT; k0 += 4) {
        const int k = k0 + 2 * half;
        v2f a, b;
        const float* ar = &gm[(tm * 16 + idx) * DFEAT + k];   // A[m,k]
        a.x = ar[0]; a.y = ar[1];
        const float* wr = &W[(tn * 16 + idx) * DFEAT + k];    // B[k,n]=W[n,k]
        b.x = wr[0]; b.y = wr[1];
        // (neg_a, A, neg_b, B, c_mod, C, reuse_a, reuse_b)
        cacc = __builtin_amdgcn_wmma_f32_16x16x4_f32(
            false, a, false, b, (short)0, cacc, false, false);
    }

    const int n = tn * 16 + idx;
    const float bn = bias[n];
    #pragma unroll
    for (int r = 0; r < 8; ++r) {
        const int m = tm * 16 + r + 8 * half;
        float v = cacc[r] + bn;
        y[m * DFEAT + n] = (v >= 0.0f) ? v : NEG_SLOPE * v;
    }
}

// ---------------------------------------------------------------------------
// Inputs (setup_inputs order):
//  0 x[N*64] f32   1 node_norm[N] f32   2 edge_norm[E] f32   3 W[64*64] f32
//  4 b[64] f32     5 src[E] i32         6 dst[E] i32         7 graph_ids[N] i32
//  8 num_graphs (device scalar, unreadable during capture -> G = out_size/64)
// ---------------------------------------------------------------------------
extern "C" void kernel_launch(void* const* d_in, const int* in_sizes, int n_in,
                              void* d_out, int out_size, void* d_ws, size_t ws_size,
                              hipStream_t stream) {
    const float* x         = (const float*)d_in[0];
    const float* node_norm = (const float*)d_in[1];
    const float* edge_norm = (const float*)d_in[2];
    const float* W         = (const float*)d_in[3];
    const float* b         = (const float*)d_in[4];
    const int*   src       = (const int*)d_in[5];
    const int*   dst       = (const int*)d_in[6];
    const int*   graph_ids = (const int*)d_in[7];

    const int N = in_sizes[1];
    const int E = in_sizes[2];
    const int G = out_size / DFEAT;            // 64

    float* g_sum  = (float*)d_ws;              // G*64 floats
    float* counts = g_sum + GMAX * DFEAT;      // G floats
    float* y      = (float*)d_out;

    // 1. zero accumulators (workspace is poisoned by the harness)
    const int nz = GMAX * DFEAT + GMAX;
    hg_zero_kernel<<<(nz + 255) / 256, 256, 0, stream>>>(g_sum, nz);

    // 2. per-graph node counts
    hg_count_kernel<<<256, 256, 0, stream>>>(graph_ids, counts, N, G);

    // 3. heavy fused edge reduction
    hg_edge_kernel<<<1024, 256, 0, stream>>>(x, node_norm, edge_norm,
                                             src, dst, graph_ids, g_sum, E, G);

    // 4. mean + WMMA GEMM + bias + LeakyReLU
    hg_gemm_kernel<<<1, 512, 0, stream>>>(g_sum, counts, W, b, y, G);
}